// ImageMPSEmbedding_25151328485537
// MI455X (gfx1250) — compile-verified
//
#include <hip/hip_runtime.h>
#include <math.h>

// ---------------------------------------------------------------------------
// ImageMPSEmbedding (angle embedding, nchan=1, inference):
//   out[p, b, 0, 0, 0] = cos(clip(x[b,0,p],0,1) * pi/2)
//   out[p, b, 0, 1, 0] = sin(clip(x[b,0,p],0,1) * pi/2)
// x: (B=1024, 1, 128, 128) fp32  ->  out: (N=16384, B=1024, 1, 2, 1) fp32
//
// Memory-bound transpose + elementwise sin/cos (HBM roofline ~8.2us for
// 192MiB @ 23.3TB/s). LDS-tiled: global reads coalesced along p via async
// global->LDS b128 copies (ASYNCcnt path), global writes coalesced along b
// as nontemporal float2 stores. LDS uses a float4 XOR swizzle so the b128
// stores stay 16B-aligned while transposed reads spread across banks.
// ---------------------------------------------------------------------------

#define HALF_PI_F 1.57079632679489661923f

#define TB 32                 // batches per tile (rows)
#define TP 64                 // pixels per tile (row length: 16 float4 cols)
#define NTHREADS 256          // 8 waves (wave32)
#define NPIX 16384            // 128*128, compile-time so addressing is shifts
#define NBATCH 1024

typedef float f32x2 __attribute__((ext_vector_type(2)));
typedef int   i32x4 __attribute__((ext_vector_type(4)));

typedef __attribute__((address_space(1))) int   g_i32;
typedef __attribute__((address_space(3))) int   l_i32;
typedef __attribute__((address_space(1))) i32x4 g_i32x4;
typedef __attribute__((address_space(3))) i32x4 l_i32x4;

#if defined(__gfx1250__) && __has_builtin(__builtin_amdgcn_global_load_async_to_lds_b128)
#define ASYNC_B128 1
#else
#define ASYNC_B128 0
#endif
#if defined(__gfx1250__) && __has_builtin(__builtin_amdgcn_global_load_async_to_lds_b32)
#define ASYNC_B32 1
#else
#define ASYNC_B32 0
#endif

__global__ __launch_bounds__(NTHREADS) void
mps_angle_embed_kernel(const float* __restrict__ x,
                       f32x2* __restrict__ out)   // out[p*NBATCH + b] = (cos, sin)
{
    // TB x TP tile, row stride 64 floats (256B). float4 column c of row r is
    // stored at column (c ^ (r & 15)) -> every store 16B aligned, transposed
    // reads land on 16 distinct bank-groups (2-way max, LDS is not the
    // bottleneck here).
    __shared__ float tile[TB * TP];

    const unsigned p0 = blockIdx.x * TP;   // pixel-tile origin
    const unsigned b0 = blockIdx.y * TB;   // batch-tile origin
    const unsigned t  = threadIdx.x;

    // ---- Phase 1: stage tile into LDS (coalesced along p) -------------------
#if ASYNC_B128
    {
        const unsigned c  = t & 15;        // float4 column within row, 0..15
        const unsigned r0 = t >> 4;        // row 0..15
#pragma unroll
        for (unsigned r = r0; r < TB; r += NTHREADS / 16) {
            const float* src = x + (unsigned)((b0 + r) * NPIX + p0 + c * 4);
            float*       dst = &tile[r * TP + ((c ^ (r & 15)) << 2)];
            __builtin_amdgcn_global_load_async_to_lds_b128(
                (g_i32x4*)src, (l_i32x4*)dst, /*offset=*/0, /*cpol=*/0);
        }
    }
#elif ASYNC_B32
    {
        const unsigned px = t & (TP - 1);
        const unsigned r0 = t / TP;
#pragma unroll
        for (unsigned r = r0; r < TB; r += NTHREADS / TP) {
            const float* src = x + (unsigned)((b0 + r) * NPIX + p0 + px);
            float*       dst = &tile[r * TP + (((px >> 2) ^ (r & 15)) << 2) + (px & 3)];
            __builtin_amdgcn_global_load_async_to_lds_b32(
                (g_i32*)src, (l_i32*)dst, 0, 0);
        }
    }
#else
    {
        const unsigned px = t & (TP - 1);
        const unsigned r0 = t / TP;
#pragma unroll
        for (unsigned r = r0; r < TB; r += NTHREADS / TP) {
            tile[r * TP + (((px >> 2) ^ (r & 15)) << 2) + (px & 3)] =
                x[(unsigned)((b0 + r) * NPIX + p0 + px)];
        }
    }
#endif

#if ASYNC_B128 || ASYNC_B32
#if __has_builtin(__builtin_amdgcn_s_wait_asynccnt)
    __builtin_amdgcn_s_wait_asynccnt(0);
#else
    asm volatile("s_wait_asynccnt 0x0" ::: "memory");
#endif
#endif
    __syncthreads();

    // ---- Phase 2: transposed read, sin/cos, coalesced NT float2 store -------
    {
        const unsigned bl  = t & (TB - 1);   // lane == batch offset (coalesced)
        const unsigned pl0 = t / TB;         // 0..7
#pragma unroll
        for (unsigned pl = pl0; pl < TP; pl += NTHREADS / TB) {
            const unsigned sc = ((pl >> 2) ^ (bl & 15)) << 2;  // un-swizzle
            float v = tile[bl * TP + sc + (pl & 3)];
            v = fminf(fmaxf(v, 0.0f), 1.0f) * HALF_PI_F;
            f32x2 cs;
            cs.x = __cosf(v);             // v_cos_f32 (TRANS)
            cs.y = __sinf(v);             // v_sin_f32 (TRANS)
            // Wave writes 32 consecutive float2 = 256B contiguous, streaming.
            __builtin_nontemporal_store(
                cs, &out[(unsigned)((p0 + pl) * NBATCH + b0 + bl)]);
        }
    }
}

extern "C" void kernel_launch(void* const* d_in, const int* in_sizes, int n_in,
                              void* d_out, int out_size, void* d_ws, size_t ws_size,
                              hipStream_t stream) {
    const float* x = (const float*)d_in[0];
    f32x2* out = (f32x2*)d_out;

    dim3 grid(NPIX / TP, NBATCH / TB);     // 256 x 32 blocks
    mps_angle_embed_kernel<<<grid, NTHREADS, 0, stream>>>(x, out);
}